// CTC_Decoding_layer_50431505990043
// MI455X (gfx1250) — compile-verified
//
#include <hip/hip_runtime.h>
#include <stdint.h>

#define BATCH 128
#define TIME 512
#define NCLS 1024
#define BLANK (NCLS - 1)
#define DUMMY 2

// ---- Phase A (argmax) geometry ----
#define WAVES_PER_BLOCK 8
#define POS_PER_WAVE 8
#define CHUNK_FLOATS 512                         // 2 KB chunk
#define CHUNKS_PER_POS (NCLS / CHUNK_FLOATS)     // 2
#define TOTAL_CHUNKS (POS_PER_WAVE * CHUNKS_PER_POS)  // 16

// Issue one 2KB chunk (512 floats) as 4 x global_load_async_to_lds_b128 per lane.
// GVS addressing: saddr = 64-bit base (SGPR pair), vaddr = 32-bit byte offset,
// vdst = 32-bit LDS byte address. Tracked on ASYNCcnt. Stream is read-once and
// larger than the 192MB L2 -> non-temporal hint.
__device__ __forceinline__ void issue_chunk_async(unsigned long long gbase,
                                                  unsigned goff, unsigned loff,
                                                  int lane) {
  const unsigned g = goff + (unsigned)lane * 16u;
  const unsigned l = loff + (unsigned)lane * 16u;
#pragma unroll
  for (int j = 0; j < 4; ++j) {
    asm volatile("global_load_async_to_lds_b128 %0, %1, %2 th:TH_LOAD_NT"
                 :
                 : "v"(l + (unsigned)(j * 512)),
                   "v"(g + (unsigned)(j * 512)),
                   "s"(gbase)
                 : "memory");
  }
}

__global__ __launch_bounds__(WAVES_PER_BLOCK * 32) void ctc_argmax_kernel(
    const float* __restrict__ in, int* __restrict__ ids) {
  __shared__ __align__(16) float stage[WAVES_PER_BLOCK][2][CHUNK_FLOATS];  // 32 KB

  const int wave = threadIdx.x >> 5;
  const int lane = threadIdx.x & 31;

  const unsigned long long gbase = (unsigned long long)(uintptr_t)in;
  // flat LDS pointer low 32 bits == workgroup-relative LDS byte address
  const unsigned loff0 = (unsigned)(uintptr_t)&stage[wave][0][0];
  const unsigned loff1 = (unsigned)(uintptr_t)&stage[wave][1][0];

  const int posBase =
      blockIdx.x * (WAVES_PER_BLOCK * POS_PER_WAVE) + wave * POS_PER_WAVE;
  const unsigned gposBase = (unsigned)posBase * (unsigned)(NCLS * 4);

  // prologue: chunk 0 -> buffer 0
  issue_chunk_async(gbase, gposBase, loff0, lane);

  float bestv = -3.402823466e+38f;
  int besti = 0;

#pragma unroll 1
  for (int k = 0; k < TOTAL_CHUNKS; ++k) {
    if (k + 1 < TOTAL_CHUNKS) {
      // buffer we are about to refill was consumed last iteration; make sure
      // its ds_load reads retired before the async engine overwrites it
      asm volatile("s_wait_dscnt 0" ::: "memory");
      const unsigned nxtLoff = ((k + 1) & 1) ? loff1 : loff0;
      issue_chunk_async(gbase, gposBase + (unsigned)(k + 1) * (CHUNK_FLOATS * 4u),
                        nxtLoff, lane);
      // 8 outstanding, wait down to 4 -> chunk k complete (loads retire in order)
      asm volatile("s_wait_asynccnt 4" ::: "memory");
    } else {
      asm volatile("s_wait_asynccnt 0" ::: "memory");
    }

    const int c = k & 1;  // chunk-within-position
    const float* bufp = (k & 1) ? &stage[wave][1][0] : &stage[wave][0][0];

#pragma unroll
    for (int j = 0; j < 4; ++j) {
      const float4 v4 = *(const float4*)(bufp + j * 128 + lane * 4);
      const int cls = c * CHUNK_FLOATS + j * 128 + lane * 4;
      if (v4.x > bestv) { bestv = v4.x; besti = cls + 0; }
      if (v4.y > bestv) { bestv = v4.y; besti = cls + 1; }
      if (v4.z > bestv) { bestv = v4.z; besti = cls + 2; }
      if (v4.w > bestv) { bestv = v4.w; besti = cls + 3; }
    }

    if (c == CHUNKS_PER_POS - 1) {
      // wave32 butterfly argmax reduce; tie -> smaller index (first occurrence)
      float v = bestv;
      int bi = besti;
#pragma unroll
      for (int m = 16; m > 0; m >>= 1) {
        const float ov = __shfl_xor(v, m, 32);
        const int oi = __shfl_xor(bi, m, 32);
        if (ov > v || (ov == v && oi < bi)) { v = ov; bi = oi; }
      }
      if (lane == 0) ids[posBase + (k >> 1)] = bi;
      bestv = -3.402823466e+38f;
      besti = 0;
    }
  }
}

// ---- Phase B: per-row stable compaction of non-blank ids (in place in d_out)
__global__ __launch_bounds__(TIME) void ctc_compact_kernel(
    int* __restrict__ io, int* __restrict__ counts) {
  __shared__ int s_scan[TIME];
  const int b = blockIdx.x;
  const int t = threadIdx.x;

  const int id = io[b * TIME + t];     // all reads before first barrier
  const int m = (id != BLANK) ? 1 : 0;
  s_scan[t] = m;
  __syncthreads();

  // Hillis-Steele inclusive scan over 512 entries
  for (int off = 1; off < TIME; off <<= 1) {
    const int v = (t >= off) ? s_scan[t - off] : 0;
    __syncthreads();
    s_scan[t] += v;
    __syncthreads();
  }
  const int incl = s_scan[t];

  if (m) io[b * TIME + (incl - 1)] = id;        // stable left-compaction
  if (t == TIME - 1) counts[b] = incl;          // row decoded length
}

// ---- Phase C: fill tail with -1 (j < max_len) / DUMMY (j >= max_len)
__global__ __launch_bounds__(TIME) void ctc_fill_kernel(
    int* __restrict__ out, const int* __restrict__ counts) {
  __shared__ int s_cnt[BATCH];
  __shared__ int s_max;
  const int b = blockIdx.x;
  const int t = threadIdx.x;

  if (t < BATCH) s_cnt[t] = counts[t];
  __syncthreads();
  if (t == 0) {
    int mx = 0;
    for (int i = 0; i < BATCH; ++i) mx = max(mx, s_cnt[i]);
    s_max = mx;
  }
  __syncthreads();

  const int cnt = s_cnt[b];
  const int mx = s_max;
  if (t >= cnt) out[b * TIME + t] = (t < mx) ? -1 : DUMMY;
}

extern "C" void kernel_launch(void* const* d_in, const int* in_sizes, int n_in,
                              void* d_out, int out_size, void* d_ws,
                              size_t ws_size, hipStream_t stream) {
  (void)in_sizes; (void)n_in; (void)out_size; (void)ws_size;
  const float* in = (const float*)d_in[0];
  int* out = (int*)d_out;          // reference output is int32; write raw int32
  int* counts = (int*)d_ws;        // 128 ints of scratch

  const int posBlocks = (BATCH * TIME) / (WAVES_PER_BLOCK * POS_PER_WAVE);  // 1024
  hipLaunchKernelGGL(ctc_argmax_kernel, dim3(posBlocks),
                     dim3(WAVES_PER_BLOCK * 32), 0, stream, in, out);
  hipLaunchKernelGGL(ctc_compact_kernel, dim3(BATCH), dim3(TIME), 0, stream,
                     out, counts);
  hipLaunchKernelGGL(ctc_fill_kernel, dim3(BATCH), dim3(TIME), 0, stream, out,
                     counts);
}